// MIPTVDepthEstimator_23991687316109
// MI455X (gfx1250) — compile-verified
//
#include <hip/hip_runtime.h>
#include <stdint.h>

#define BATCH 16
#define H 512
#define W 512
#define HW (H * W)
#define HALO 15              // MIP_K=31 -> radius 15
#define NBINS 4096
#define NEG_HUGE (-3.402823466e+38f)

// ---------------------------------------------------------------------------
// CDNA5 async global->LDS load (per-lane): 4 bytes from the global address in vaddr to the LDS address in vdst.
// Tracked by ASYNCcnt; we wait with s_wait_asynccnt 0 before the barrier.
// ---------------------------------------------------------------------------
__device__ __forceinline__ unsigned lds_off_u32(const void* p) {
  // generic pointer to __shared__: low 32 bits are the LDS byte offset
  return (unsigned)(uintptr_t)p;
}

__device__ __forceinline__ void async_ld_b32(void* lds_dst, const void* gsrc) {
  unsigned lo = lds_off_u32(lds_dst);
  uint64_t ga = (uint64_t)(uintptr_t)gsrc;
  asm volatile("global_load_async_to_lds_b32 %0, %1, off"
               :: "v"(lo), "v"(ga) : "memory");
}

__device__ __forceinline__ void wait_async0() {
  asm volatile("s_wait_asynccnt 0" ::: "memory");
}

__device__ __forceinline__ float4 ld4(const float* p) {
  return *(const float4*)p;
}
__device__ __forceinline__ void st4(float* p, float4 v) {
  *(float4*)p = v;
}

// ---------------------------------------------------------------------------
// Init: zero histograms, seed per-batch stats (min=0xFFFFFFFF bits, max=0)
// statsU layout per batch (8 u32): [0]=xmin [1]=xmax [2]=lo [3]=hi
//                                  [4]=gmin [5]=gmax [6]=sigma [7]=pad
// ---------------------------------------------------------------------------
__global__ __launch_bounds__(256) void k_init(unsigned* hist, unsigned* statsU) {
  int i = blockIdx.x * 256 + threadIdx.x;
  if (i < 2 * BATCH * NBINS) hist[i] = 0u;
  if (i < BATCH) {
    unsigned* s = statsU + i * 8;
    s[0] = 0xFFFFFFFFu; s[1] = 0u; s[2] = 0u; s[3] = 0u;
    s[4] = 0xFFFFFFFFu; s[5] = 0u; s[6] = 0u; s[7] = 0u;
  }
}

// ---------------------------------------------------------------------------
// Pass 1: horizontal 31-tap running max per (b,c,row). Async halo tile in LDS.
// grid: (W/256, H, B*3), block 256
// ---------------------------------------------------------------------------
__global__ __launch_bounds__(256) void k_rowmax(const float* __restrict__ in,
                                                float* __restrict__ out) {
  __shared__ float s[256 + 2 * HALO];
  const int t  = threadIdx.x;
  const int x0 = blockIdx.x * 256;
  const int row = blockIdx.y;
  const int zc  = blockIdx.z;                    // b*3 + c
  const float* rowp = in + ((size_t)zc * H + row) * W;

  for (int idx = t; idx < 256 + 2 * HALO; idx += 256) {
    int gx = x0 - HALO + idx;
    if (gx >= 0 && gx < W) {
      async_ld_b32(&s[idx], rowp + gx);
    } else {
      s[idx] = NEG_HUGE;
    }
  }
  wait_async0();
  __syncthreads();

  float m = s[t];
#pragma unroll
  for (int k = 1; k <= 2 * HALO; ++k) m = fmaxf(m, s[t + k]);
  out[((size_t)zc * H + row) * W + x0 + t] = m;
}

// ---------------------------------------------------------------------------
// Pass 2: vertical 31-tap max for R,G,B row-max planes; fuse
//   x = |Rmax - max(Gmax,Bmax)| + R, plus per-batch min/max of x (uint atomics,
//   valid because x >= 0). Tile 32x8 outputs, 32x38 LDS per channel (async).
// grid: (W/32, H/8, B), block 256 (32x8)
// ---------------------------------------------------------------------------
__global__ __launch_bounds__(256) void k_colmax_x(const float* __restrict__ in,
                                                  const float* __restrict__ rowmax,
                                                  float* __restrict__ xout,
                                                  unsigned* __restrict__ statsU) {
  const int TILE_W = 32, TILE_H = 8, LDS_H = TILE_H + 2 * HALO;   // 38
  __shared__ float sR[LDS_H * TILE_W];
  __shared__ float sG[LDS_H * TILE_W];
  __shared__ float sB[LDS_H * TILE_W];
  __shared__ float redmin[256];
  __shared__ float redmax[256];

  const int t  = threadIdx.x;
  const int tx = t & 31;
  const int ty = t >> 5;
  const int x0 = blockIdx.x * TILE_W;
  const int y0 = blockIdx.y * TILE_H;
  const int b  = blockIdx.z;

  for (int c = 0; c < 3; ++c) {
    float* sm = (c == 0) ? sR : (c == 1) ? sG : sB;
    const float* src = rowmax + ((size_t)b * 3 + c) * (size_t)H * W;
    for (int idx = t; idx < LDS_H * TILE_W; idx += 256) {
      int ly = idx >> 5, lx = idx & 31;
      int gy = y0 - HALO + ly;
      if (gy >= 0 && gy < H) {
        async_ld_b32(&sm[idx], src + (size_t)gy * W + x0 + lx);
      } else {
        sm[idx] = NEG_HUGE;
      }
    }
  }
  wait_async0();
  __syncthreads();

  float rM = NEG_HUGE, gM = NEG_HUGE, bM = NEG_HUGE;
#pragma unroll
  for (int k = 0; k < 2 * HALO + 1; ++k) {
    int off = (ty + k) * TILE_W + tx;
    rM = fmaxf(rM, sR[off]);
    gM = fmaxf(gM, sG[off]);
    bM = fmaxf(bM, sB[off]);
  }
  const int y = y0 + ty, x = x0 + tx;
  float R  = in[(((size_t)b * 3 + 0) * H + y) * W + x];
  float xv = fabsf(rM - fmaxf(gM, bM)) + R;
  xout[((size_t)b * HW) + (size_t)y * W + x] = xv;

  // block min/max reduction -> one atomic pair per block
  redmin[t] = xv; redmax[t] = xv;
  __syncthreads();
  for (int sft = 128; sft > 0; sft >>= 1) {
    if (t < sft) {
      redmin[t] = fminf(redmin[t], redmin[t + sft]);
      redmax[t] = fmaxf(redmax[t], redmax[t + sft]);
    }
    __syncthreads();
  }
  if (t == 0) {
    atomicMin(&statsU[b * 8 + 0], __float_as_uint(redmin[0]));
    atomicMax(&statsU[b * 8 + 1], __float_as_uint(redmax[0]));
  }
}

// ---------------------------------------------------------------------------
// Per-batch histogram of x over [xmin, xmax] (LDS hist, ds_add atomics).
// grid: (16 blocks-per-batch, B), block 256
// ---------------------------------------------------------------------------
__global__ __launch_bounds__(256) void k_hist_x(const float* __restrict__ xin,
                                                const unsigned* __restrict__ statsU,
                                                unsigned* __restrict__ hist) {
  __shared__ unsigned lh[NBINS];
  const int t = threadIdx.x;
  const int b = blockIdx.y;
  for (int i = t; i < NBINS; i += 256) lh[i] = 0u;
  __syncthreads();

  const float mn = __uint_as_float(statsU[b * 8 + 0]);
  const float mx = __uint_as_float(statsU[b * 8 + 1]);
  const float scale = (float)NBINS / fmaxf(mx - mn, 1e-12f);
  const float* base = xin + (size_t)b * HW;

  for (int p = blockIdx.x * 256 + t; p < HW; p += 16 * 256) {
    float v = base[p];
    int bin = (int)((v - mn) * scale);
    bin = min(max(bin, 0), NBINS - 1);
    atomicAdd(&lh[bin], 1u);
  }
  __syncthreads();
  for (int i = t; i < NBINS; i += 256)
    if (lh[i]) atomicAdd(&hist[(size_t)b * NBINS + i], lh[i]);
}

// ---------------------------------------------------------------------------
// Per-batch histogram of grad_mag(N_hat) over [gmin, gmax] (recomputed).
// ---------------------------------------------------------------------------
__device__ __forceinline__ float grad_mag_at(const float* __restrict__ nh, int p) {
  const int x = p & (W - 1);
  const int y = p >> 9;            // W = 512
  float c  = nh[p];
  float dx = (x < W - 1) ? nh[p + 1] - c : 0.f;
  float dy = (y < H - 1) ? nh[p + W] - c : 0.f;
  return sqrtf(dx * dx + dy * dy + 1e-6f);
}

__global__ __launch_bounds__(256) void k_gradminmax(const float* __restrict__ nhat,
                                                    unsigned* __restrict__ statsU) {
  __shared__ float redmin[256];
  __shared__ float redmax[256];
  const int t = threadIdx.x;
  const int b = blockIdx.y;
  const int p = blockIdx.x * 256 + t;
  const float* nh = nhat + (size_t)b * HW;
  float g = grad_mag_at(nh, p);
  redmin[t] = g; redmax[t] = g;
  __syncthreads();
  for (int sft = 128; sft > 0; sft >>= 1) {
    if (t < sft) {
      redmin[t] = fminf(redmin[t], redmin[t + sft]);
      redmax[t] = fmaxf(redmax[t], redmax[t + sft]);
    }
    __syncthreads();
  }
  if (t == 0) {
    atomicMin(&statsU[b * 8 + 4], __float_as_uint(redmin[0]));
    atomicMax(&statsU[b * 8 + 5], __float_as_uint(redmax[0]));
  }
}

__global__ __launch_bounds__(256) void k_hist_g(const float* __restrict__ nhat,
                                                const unsigned* __restrict__ statsU,
                                                unsigned* __restrict__ hist) {
  __shared__ unsigned lh[NBINS];
  const int t = threadIdx.x;
  const int b = blockIdx.y;
  for (int i = t; i < NBINS; i += 256) lh[i] = 0u;
  __syncthreads();

  const float mn = __uint_as_float(statsU[b * 8 + 4]);
  const float mx = __uint_as_float(statsU[b * 8 + 5]);
  const float scale = (float)NBINS / fmaxf(mx - mn, 1e-12f);
  const float* nh = nhat + (size_t)b * HW;

  for (int p = blockIdx.x * 256 + t; p < HW; p += 16 * 256) {
    float g = grad_mag_at(nh, p);
    int bin = (int)((g - mn) * scale);
    bin = min(max(bin, 0), NBINS - 1);
    atomicAdd(&lh[bin], 1u);
  }
  __syncthreads();
  for (int i = t; i < NBINS; i += 256)
    if (lh[i]) atomicAdd(&hist[(size_t)b * NBINS + i], lh[i]);
}

// ---------------------------------------------------------------------------
// Quantile from histogram: one block per batch. Block-wide inclusive scan of
// 256 per-thread partial sums (16 bins each), then in-bin linear interp.
// ---------------------------------------------------------------------------
__global__ __launch_bounds__(256) void k_quantile(const unsigned* __restrict__ hist,
                                                  unsigned* __restrict__ statsU,
                                                  int minSlot, int maxSlot,
                                                  float q0, int outSlot0,
                                                  float q1, int outSlot1) {
  __shared__ unsigned psum[256];
  const int t = threadIdx.x;
  const int b = blockIdx.x;
  const int PER = NBINS / 256;     // 16
  const unsigned* h = hist + (size_t)b * NBINS;

  unsigned local[PER];
  unsigned sum = 0;
#pragma unroll
  for (int i = 0; i < PER; ++i) { local[i] = h[t * PER + i]; sum += local[i]; }
  psum[t] = sum;
  __syncthreads();
  for (int off = 1; off < 256; off <<= 1) {
    unsigned v = (t >= off) ? psum[t - off] : 0u;
    __syncthreads();
    psum[t] += v;
    __syncthreads();
  }
  const unsigned total = psum[255];
  const unsigned excl  = psum[t] - sum;

  const float mn = __uint_as_float(statsU[b * 8 + minSlot]);
  const float mx = __uint_as_float(statsU[b * 8 + maxSlot]);
  const float binw = fmaxf(mx - mn, 1e-12f) / (float)NBINS;

  float qs[2]   = {q0, q1};
  int   outs[2] = {outSlot0, outSlot1};
#pragma unroll
  for (int qi = 0; qi < 2; ++qi) {
    float tgt = qs[qi] * (float)(total - 1);
    unsigned cum = excl;
#pragma unroll
    for (int i = 0; i < PER; ++i) {
      unsigned cnt = local[i];
      if (cnt > 0 && (float)cum <= tgt && tgt < (float)cum + (float)cnt) {
        float frac = (tgt - (float)cum + 0.5f) / (float)cnt;
        frac = fminf(fmaxf(frac, 0.f), 1.f);
        float val = mn + ((float)(t * PER + i) + frac) * binw;
        statsU[b * 8 + outs[qi]] = __float_as_uint(val);
      }
      cum += cnt;
    }
  }
}

// ---------------------------------------------------------------------------
// N_hat = clip((x - lo)/(hi - lo + eps), 0, 1).  float4: grid (HW/1024, B)
// ---------------------------------------------------------------------------
__global__ __launch_bounds__(256) void k_nhat(const float* __restrict__ xin,
                                              const unsigned* __restrict__ statsU,
                                              float* __restrict__ nh) {
  const int b = blockIdx.y;
  const int p = (blockIdx.x * 256 + threadIdx.x) * 4;
  const float lo = __uint_as_float(statsU[b * 8 + 2]);
  const float hi = __uint_as_float(statsU[b * 8 + 3]);
  const float inv = 1.0f / (hi - lo + 1e-6f);
  const size_t o = (size_t)b * HW + p;
  float4 v = ld4(xin + o);
  v.x = fminf(fmaxf((v.x - lo) * inv, 0.f), 1.f);
  v.y = fminf(fmaxf((v.y - lo) * inv, 0.f), 1.f);
  v.z = fminf(fmaxf((v.z - lo) * inv, 0.f), 1.f);
  v.w = fminf(fmaxf((v.w - lo) * inv, 0.f), 1.f);
  st4(nh + o, v);
}

// ---------------------------------------------------------------------------
// Wx, Wy from grad of N_hat + sigma; init u = ubar = N_hat, px = py = 0.
// float4: one lane = 4 contiguous pixels of one row (W % 4 == 0).
// ---------------------------------------------------------------------------
__global__ __launch_bounds__(256) void k_wxy_init(const float* __restrict__ nhat,
                                                  const unsigned* __restrict__ statsU,
                                                  float* __restrict__ wx, float* __restrict__ wy,
                                                  float* __restrict__ u,  float* __restrict__ ubar,
                                                  float* __restrict__ px, float* __restrict__ py) {
  const int b = blockIdx.y;
  const int p = (blockIdx.x * 256 + threadIdx.x) * 4;
  const int x = p & (W - 1);
  const int y = p >> 9;
  const float sigma = fmaxf(__uint_as_float(statsU[b * 8 + 6]), 1e-6f);
  const float inv_sigma = 1.0f / sigma;
  const size_t o = (size_t)b * HW + p;

  float4 c = ld4(nhat + o);
  float cr = (x + 4 < W) ? nhat[o + 4] : c.w;       // right edge -> dx = 0
  float4 d = (y < H - 1) ? ld4(nhat + o + W) : c;   // bottom edge -> dy = 0

  float4 dx = make_float4(c.y - c.x, c.z - c.y, c.w - c.z, cr - c.w);
  float4 dy = make_float4(d.x - c.x, d.y - c.y, d.z - c.z, d.w - c.w);

  float4 wx4 = make_float4(1.0f + 10.0f * __expf(-fabsf(dx.x) * inv_sigma),
                           1.0f + 10.0f * __expf(-fabsf(dx.y) * inv_sigma),
                           1.0f + 10.0f * __expf(-fabsf(dx.z) * inv_sigma),
                           1.0f + 10.0f * __expf(-fabsf(dx.w) * inv_sigma));
  float4 wy4 = make_float4(1.0f + 10.0f * __expf(-fabsf(dy.x) * inv_sigma),
                           1.0f + 10.0f * __expf(-fabsf(dy.y) * inv_sigma),
                           1.0f + 10.0f * __expf(-fabsf(dy.z) * inv_sigma),
                           1.0f + 10.0f * __expf(-fabsf(dy.w) * inv_sigma));
  st4(wx + o, wx4);
  st4(wy + o, wy4);
  st4(u + o, c);
  st4(ubar + o, c);
  float4 z = make_float4(0.f, 0.f, 0.f, 0.f);
  st4(px + o, z);
  st4(py + o, z);
}

// ---------------------------------------------------------------------------
// PD step 1: dual update. px,py clipped to +/- 0.15*W. float4 per lane.
// ---------------------------------------------------------------------------
__global__ __launch_bounds__(256) void k_pd1(const float* __restrict__ ubar,
                                             const float* __restrict__ wx,
                                             const float* __restrict__ wy,
                                             float* __restrict__ px,
                                             float* __restrict__ py) {
  const int b = blockIdx.y;
  const int p = (blockIdx.x * 256 + threadIdx.x) * 4;
  const int x = p & (W - 1);
  const int y = p >> 9;
  const size_t o = (size_t)b * HW + p;

  float4 ub = ld4(ubar + o);
  float ubr = (x + 4 < W) ? ubar[o + 4] : ub.w;
  float4 ud = (y < H - 1) ? ld4(ubar + o + W) : ub;

  float4 dx = make_float4(ub.y - ub.x, ub.z - ub.y, ub.w - ub.z, ubr - ub.w);
  float4 dy = make_float4(ud.x - ub.x, ud.y - ub.y, ud.z - ub.z, ud.w - ub.w);

  float4 wx4 = ld4(wx + o);
  float4 wy4 = ld4(wy + o);
  float4 px4 = ld4(px + o);
  float4 py4 = ld4(py + o);

  float ax;
  ax = 0.15f * wx4.x; px4.x = fminf(fmaxf(px4.x + 0.125f * dx.x, -ax), ax);
  ax = 0.15f * wx4.y; px4.y = fminf(fmaxf(px4.y + 0.125f * dx.y, -ax), ax);
  ax = 0.15f * wx4.z; px4.z = fminf(fmaxf(px4.z + 0.125f * dx.z, -ax), ax);
  ax = 0.15f * wx4.w; px4.w = fminf(fmaxf(px4.w + 0.125f * dx.w, -ax), ax);

  float ay;
  ay = 0.15f * wy4.x; py4.x = fminf(fmaxf(py4.x + 0.125f * dy.x, -ay), ay);
  ay = 0.15f * wy4.y; py4.y = fminf(fmaxf(py4.y + 0.125f * dy.y, -ay), ay);
  ay = 0.15f * wy4.z; py4.z = fminf(fmaxf(py4.z + 0.125f * dy.z, -ay), ay);
  ay = 0.15f * wy4.w; py4.w = fminf(fmaxf(py4.w + 0.125f * dy.w, -ay), ay);

  st4(px + o, px4);
  st4(py + o, py4);
}

// ---------------------------------------------------------------------------
// PD step 2: primal update + over-relaxation. float4 per lane.
// ---------------------------------------------------------------------------
__global__ __launch_bounds__(256) void k_pd2(const float* __restrict__ nhat,
                                             const float* __restrict__ px,
                                             const float* __restrict__ py,
                                             float* __restrict__ u,
                                             float* __restrict__ ubar) {
  const int b = blockIdx.y;
  const int p = (blockIdx.x * 256 + threadIdx.x) * 4;
  const int x = p & (W - 1);
  const int y = p >> 9;
  const size_t o = (size_t)b * HW + p;

  float4 px4 = ld4(px + o);
  float  pxl = (x > 0) ? px[o - 1] : 0.f;
  float4 py4 = ld4(py + o);
  float4 pyu = (y > 0) ? ld4(py + o - W) : make_float4(0.f, 0.f, 0.f, 0.f);

  float4 div;
  div.x = (px4.x - pxl)   + (py4.x - pyu.x);
  div.y = (px4.y - px4.x) + (py4.y - pyu.y);
  div.z = (px4.z - px4.y) + (py4.z - pyu.z);
  div.w = (px4.w - px4.z) + (py4.w - pyu.w);

  float4 u4 = ld4(u + o);
  float4 n4 = ld4(nhat + o);
  const float inv = 1.0f / 1.125f;

  float4 un, ub;
  un.x = (u4.x + 0.125f * div.x + 0.125f * n4.x) * inv; ub.x = 2.f * un.x - u4.x;
  un.y = (u4.y + 0.125f * div.y + 0.125f * n4.y) * inv; ub.y = 2.f * un.y - u4.y;
  un.z = (u4.z + 0.125f * div.z + 0.125f * n4.z) * inv; ub.z = 2.f * un.z - u4.z;
  un.w = (u4.w + 0.125f * div.w + 0.125f * n4.w) * inv; ub.w = 2.f * un.w - u4.w;

  st4(ubar + o, ub);
  st4(u + o, un);
}

__global__ __launch_bounds__(256) void k_out(const float* __restrict__ u,
                                             float* __restrict__ out) {
  const int b = blockIdx.y;
  const int p = (blockIdx.x * 256 + threadIdx.x) * 4;
  const size_t o = (size_t)b * HW + p;
  float4 v = ld4(u + o);
  v.x = fminf(fmaxf(v.x, 0.f), 1.f);
  v.y = fminf(fmaxf(v.y, 0.f), 1.f);
  v.z = fminf(fmaxf(v.z, 0.f), 1.f);
  v.w = fminf(fmaxf(v.w, 0.f), 1.f);
  st4(out + o, v);
}

// ---------------------------------------------------------------------------
// Host-side launch sequence (graph-capture safe: only kernel launches).
// ---------------------------------------------------------------------------
extern "C" void kernel_launch(void* const* d_in, const int* in_sizes, int n_in,
                              void* d_out, int out_size, void* d_ws, size_t ws_size,
                              hipStream_t stream) {
  (void)in_sizes; (void)n_in; (void)out_size; (void)ws_size;
  const float* Iy = (const float*)d_in[0];
  float* out = (float*)d_out;
  float* ws  = (float*)d_ws;

  size_t off = 0;
  float* rowmax = ws + off; off += (size_t)BATCH * 3 * HW;
  float* xbuf   = ws + off; off += (size_t)BATCH * HW;
  float* nhat   = ws + off; off += (size_t)BATCH * HW;
  float* wx     = ws + off; off += (size_t)BATCH * HW;
  float* wy     = ws + off; off += (size_t)BATCH * HW;
  float* u      = ws + off; off += (size_t)BATCH * HW;
  float* ubar   = ws + off; off += (size_t)BATCH * HW;
  float* px     = ws + off; off += (size_t)BATCH * HW;
  float* py     = ws + off; off += (size_t)BATCH * HW;
  unsigned* hist   = (unsigned*)(ws + off); off += (size_t)2 * BATCH * NBINS;
  unsigned* statsU = (unsigned*)(ws + off); off += (size_t)BATCH * 8;
  unsigned* histX = hist;
  unsigned* histG = hist + (size_t)BATCH * NBINS;

  const dim3 blk(256);
  const dim3 gEw(HW / 256, BATCH);          // scalar elementwise grid (1024, 16)
  const dim3 gV4(HW / 1024, BATCH);         // float4 elementwise grid (256, 16)

  k_init<<<dim3((2 * BATCH * NBINS + 255) / 256), blk, 0, stream>>>(hist, statsU);

  // 31x31 max-pool, separable; async global->LDS halo tiles
  k_rowmax<<<dim3(W / 256, H, BATCH * 3), blk, 0, stream>>>(Iy, rowmax);
  k_colmax_x<<<dim3(W / 32, H / 8, BATCH), blk, 0, stream>>>(Iy, rowmax, xbuf, statsU);

  // quantiles of x -> lo (1%), hi (99%)
  k_hist_x<<<dim3(16, BATCH), blk, 0, stream>>>(xbuf, statsU, histX);
  k_quantile<<<dim3(BATCH), blk, 0, stream>>>(histX, statsU, 0, 1, 0.01f, 2, 0.99f, 3);
  k_nhat<<<gV4, blk, 0, stream>>>(xbuf, statsU, nhat);

  // median of |grad N_hat| -> sigma
  k_gradminmax<<<gEw, blk, 0, stream>>>(nhat, statsU);
  k_hist_g<<<dim3(16, BATCH), blk, 0, stream>>>(nhat, statsU, histG);
  k_quantile<<<dim3(BATCH), blk, 0, stream>>>(histG, statsU, 4, 5, 0.5f, 6, 0.5f, 7);

  // weights + PD state init
  k_wxy_init<<<gV4, blk, 0, stream>>>(nhat, statsU, wx, wy, u, ubar, px, py);

  // 30 primal-dual iterations; full state (~117 MB) is L2-resident on MI455X
  for (int it = 0; it < 30; ++it) {
    k_pd1<<<gV4, blk, 0, stream>>>(ubar, wx, wy, px, py);
    k_pd2<<<gV4, blk, 0, stream>>>(nhat, px, py, u, ubar);
  }

  k_out<<<gV4, blk, 0, stream>>>(u, out);
}